// GatedGCNLayer_42537356100034
// MI455X (gfx1250) — compile-verified
//
#include <hip/hip_runtime.h>
#include <cstdint>
#include <cstddef>

typedef __attribute__((ext_vector_type(2))) float v2f;
typedef __attribute__((ext_vector_type(8))) float v8f;

#define GD 128        // feature dim
#define GN 50000      // nodes
#define GE 600000     // edges
#define WPB 4         // waves per block in GEMM kernels (wave32)
#define TPB (WPB * 32)

// ---------------------------------------------------------------------------
// gfx1250 async global->LDS copy (ASYNCcnt path), GV addressing mode:
//   vdst = LDS byte address, vaddr = 64-bit global address.
// ---------------------------------------------------------------------------
__device__ __forceinline__ void async_copy_b128(uint32_t lds_addr,
                                                const void* gptr) {
  asm volatile("global_load_async_to_lds_b128 %0, %1, off"
               :: "v"(lds_addr), "v"((uint64_t)(uintptr_t)gptr)
               : "memory");
}
__device__ __forceinline__ void wait_asynccnt0() {
  asm volatile("s_wait_asynccnt 0x0" ::: "memory");
}

// ---------------------------------------------------------------------------
// Stage W (128x128 f32, row-major) into LDS:
//  1) async-copy 64KB verbatim into lWraw (global_load_async_to_lds_b128)
//  2) swizzle to paired-K layout: lWp[k2*128 + c] = {W[2k2][c], W[2k2+1][c]}
// so each WMMA B fragment is ONE aligned ds_load_b64, conflict-free.
// ---------------------------------------------------------------------------
__device__ __forceinline__ void stage_w(const float* __restrict__ W,
                                        float* lWraw, v2f* lWp) {
  for (int i = threadIdx.x * 4; i < GD * GD; i += (int)blockDim.x * 4)
    async_copy_b128((uint32_t)(uintptr_t)&lWraw[i], W + i);
  wait_asynccnt0();
  __syncthreads();
  for (int p = threadIdx.x; p < (GD / 2) * GD; p += (int)blockDim.x) {
    const int k2 = p >> 7, c = p & (GD - 1);
    lWp[p] = (v2f){lWraw[(2 * k2) * GD + c], lWraw[(2 * k2 + 1) * GD + c]};
  }
  __syncthreads();
}

// ---------------------------------------------------------------------------
// Wave-level 16x128 GEMM tile: acc += X[rowBase:rowBase+16, :] @ W
// V_WMMA_F32_16X16X4_F32.  A 16x4: lane&15 = row M, VGPR pair = K(2*hi)+{0,1}
// (one global b64).  B 4x16: paired-K LDS -> one ds_load_b64 per fragment.
// C/D 16x16: VGPR v holds M = v + 8*hi, N = lane&15.
// ---------------------------------------------------------------------------
__device__ __forceinline__ void wmma_tile_16x128(const float* __restrict__ X,
                                                 int rowBase,
                                                 const v2f* __restrict__ lWp,
                                                 v8f acc[8]) {
  const int lane = threadIdx.x & 31;
  const int l15  = lane & 15;
  const int hi   = lane >> 4;
  const float* xr = X + (size_t)(rowBase + l15) * GD;
  for (int k0 = 0; k0 < GD; k0 += 4) {
    const v2f a = *(const v2f*)(xr + k0 + 2 * hi);       // global b64
    const v2f* bp = lWp + ((k0 >> 1) + hi) * GD + l15;   // paired-K row
#pragma unroll
    for (int nt = 0; nt < 8; ++nt) {
      const v2f b = bp[nt * 16];                          // ds_load_b64
      acc[nt] = __builtin_amdgcn_wmma_f32_16x16x4_f32(
          false, a, false, b, (short)0, acc[nt], false, false);
    }
  }
}

// ---------------------------------------------------------------------------
// K1: per-node GEMMs  hA=h@Aw+Ab, hB=h@Bw+Bb, hU=h@Uw+Ub, hV=h@Vw+Vb
// ---------------------------------------------------------------------------
__global__ __launch_bounds__(TPB) void node_gemm4(
    const float* __restrict__ hmat,
    const float* __restrict__ Aw, const float* __restrict__ Ab,
    const float* __restrict__ Bw, const float* __restrict__ Bb,
    const float* __restrict__ Uw, const float* __restrict__ Ub,
    const float* __restrict__ Vw, const float* __restrict__ Vb,
    float* __restrict__ hA, float* __restrict__ hB,
    float* __restrict__ hU, float* __restrict__ hV) {
  __shared__ float lWraw[GD * GD];            // 64 KB async landing zone
  __shared__ v2f lWp[(GD / 2) * GD];          // 64 KB paired-K layout
  const int wave = threadIdx.x >> 5;
  const int lane = threadIdx.x & 31;
  const int l15 = lane & 15, hi = lane >> 4;
  const int tile = blockIdx.x * WPB + wave;
  const int nTiles = GN / 16;  // 3125
  const float* Ws[4]   = {Aw, Bw, Uw, Vw};
  const float* bias[4] = {Ab, Bb, Ub, Vb};
  float* outs[4] = {hA, hB, hU, hV};
  for (int m = 0; m < 4; ++m) {
    __syncthreads();
    stage_w(Ws[m], lWraw, lWp);
    if (tile < nTiles) {  // wave-uniform branch: EXEC stays all-ones
      v8f acc[8];
#pragma unroll
      for (int i = 0; i < 8; ++i) acc[i] = (v8f){0, 0, 0, 0, 0, 0, 0, 0};
      wmma_tile_16x128(hmat, tile * 16, lWp, acc);
      float* o = outs[m];
      const float* bv = bias[m];
#pragma unroll
      for (int nt = 0; nt < 8; ++nt) {
        const int col = nt * 16 + l15;
        const float b = bv[col];
#pragma unroll
        for (int v = 0; v < 8; ++v)
          o[(size_t)(tile * 16 + v + 8 * hi) * GD + col] = acc[nt][v] + b;
      }
    }
  }
}

// ---------------------------------------------------------------------------
// K2: eta = e@Cw + Cb + hA[senders] + hB[receivers]  -> etaOut (d_out e slot)
//     + per-channel sum / sumsq for edge BatchNorm (LDS reduce + global atomic)
// ---------------------------------------------------------------------------
__global__ __launch_bounds__(TPB) void edge_gemm(
    const float* __restrict__ e, const float* __restrict__ Cw,
    const float* __restrict__ Cb,
    const float* __restrict__ hA, const float* __restrict__ hB,
    const int* __restrict__ senders, const int* __restrict__ receivers,
    float* __restrict__ etaOut, float* __restrict__ esum,
    float* __restrict__ esumsq) {
  __shared__ float lWraw[GD * GD];
  __shared__ v2f lWp[(GD / 2) * GD];
  __shared__ float bsum[GD];
  __shared__ float bsq[GD];
  if (threadIdx.x < GD) { bsum[threadIdx.x] = 0.f; bsq[threadIdx.x] = 0.f; }
  stage_w(Cw, lWraw, lWp);
  const int wave = threadIdx.x >> 5;
  const int lane = threadIdx.x & 31;
  const int l15 = lane & 15, hi = lane >> 4;
  const int tile = blockIdx.x * WPB + wave;
  if (tile < GE / 16) {  // 37500 tiles, exact
    v8f acc[8];
#pragma unroll
    for (int i = 0; i < 8; ++i) acc[i] = (v8f){0, 0, 0, 0, 0, 0, 0, 0};
    wmma_tile_16x128(e, tile * 16, lWp, acc);
    int sidx[8], ridx[8];
#pragma unroll
    for (int v = 0; v < 8; ++v) {
      const int row = tile * 16 + v + 8 * hi;
      sidx[v] = senders[row];
      ridx[v] = receivers[row];
    }
#pragma unroll
    for (int nt = 0; nt < 8; ++nt) {
      const int col = nt * 16 + l15;
      const float cb = Cb[col];
      float s = 0.f, q = 0.f;
#pragma unroll
      for (int v = 0; v < 8; ++v) {
        const int row = tile * 16 + v + 8 * hi;
        const float val = acc[nt][v] + cb + hA[(size_t)sidx[v] * GD + col] +
                          hB[(size_t)ridx[v] * GD + col];
        etaOut[(size_t)row * GD + col] = val;
        s += val;
        q += val * val;
      }
      atomicAdd(&bsum[col], s);  // ds_add_f32 block reduction
      atomicAdd(&bsq[col], q);
    }
  }
  __syncthreads();
  if (threadIdx.x < GD) {
    atomicAdd(&esum[threadIdx.x], bsum[threadIdx.x]);
    atomicAdd(&esumsq[threadIdx.x], bsq[threadIdx.x]);
  }
}

// ---------------------------------------------------------------------------
// BN fold: g = scale*rsqrt(var+eps), b = offset - mean*g   (128 threads)
// ---------------------------------------------------------------------------
__global__ void bn_finalize(const float* __restrict__ sum,
                            const float* __restrict__ sumsq,
                            const float* __restrict__ scale,
                            const float* __restrict__ offset, float invCount,
                            float* __restrict__ gb) {
  const int c = threadIdx.x;
  const float mean = sum[c] * invCount;
  const float var = sumsq[c] * invCount - mean * mean;
  const float g = scale[c] * rsqrtf(var + 1e-5f);
  gb[c] = g;
  gb[GD + c] = offset[c] - mean * g;
}

// ---------------------------------------------------------------------------
// K3: e_new = e + relu(BN(eta));  w = sigmoid(e_new);
//     w_sum[s] += w;  agg[s] += hV[r] * w       (L2 float atomics)
// ---------------------------------------------------------------------------
__global__ void edge_apply(const float* __restrict__ e,
                           const float* __restrict__ ebn,
                           const int* __restrict__ senders,
                           const int* __restrict__ receivers,
                           const float* __restrict__ hV,
                           float* __restrict__ eio,  // in: eta, out: e_new
                           float* __restrict__ w_sum, float* __restrict__ agg) {
  const int idx = blockIdx.x * blockDim.x + threadIdx.x;
  const int row = idx >> 7, col = idx & 127;
  const float eta = eio[idx];
  const float feat = fmaxf(fmaf(eta, ebn[col], ebn[GD + col]), 0.f);
  const float en = e[idx] + feat;
  eio[idx] = en;
  const float w = 1.f / (1.f + expf(-en));
  const int s = senders[row], r = receivers[row];
  atomicAdd(&w_sum[(size_t)s * GD + col], w);
  atomicAdd(&agg[(size_t)s * GD + col], hV[(size_t)r * GD + col] * w);
}

// ---------------------------------------------------------------------------
// K4: pre = hU + agg/(w_sum+1e-6) -> d_out h slot; per-channel BN stats.
// Grid-stride with stride % 128 == 0 so each thread's channel is fixed.
// ---------------------------------------------------------------------------
__global__ void node_pre(const float* __restrict__ hU,
                         const float* __restrict__ agg,
                         const float* __restrict__ w_sum,
                         float* __restrict__ pre, float* __restrict__ nsum,
                         float* __restrict__ nsumsq) {
  const int total = GN * GD;
  const int stride = gridDim.x * blockDim.x;  // 1024*256, multiple of 128
  const int idx0 = blockIdx.x * blockDim.x + threadIdx.x;
  const int col = idx0 & 127;
  float s = 0.f, q = 0.f;
  for (int idx = idx0; idx < total; idx += stride) {
    const float v = hU[idx] + agg[idx] / (w_sum[idx] + 1e-6f);
    pre[idx] = v;
    s += v;
    q += v * v;
  }
  atomicAdd(&nsum[col], s);
  atomicAdd(&nsumsq[col], q);
}

// ---------------------------------------------------------------------------
// K5: h_new = h + relu(BN(pre))
// ---------------------------------------------------------------------------
__global__ void node_apply(const float* __restrict__ h,
                           const float* __restrict__ nbn,
                           float* __restrict__ hio) {
  const int idx = blockIdx.x * blockDim.x + threadIdx.x;
  const int col = idx & 127;
  const float pre = hio[idx];
  const float f = fmaxf(fmaf(pre, nbn[col], nbn[GD + col]), 0.f);
  hio[idx] = h[idx] + f;
}

// ---------------------------------------------------------------------------
extern "C" void kernel_launch(void* const* d_in, const int* in_sizes, int n_in,
                              void* d_out, int out_size, void* d_ws,
                              size_t ws_size, hipStream_t stream) {
  const float* h  = (const float*)d_in[0];
  const float* e  = (const float*)d_in[1];
  const int* senders   = (const int*)d_in[2];
  const int* receivers = (const int*)d_in[3];
  const float* Aw = (const float*)d_in[4];
  const float* Ab = (const float*)d_in[5];
  const float* Bw = (const float*)d_in[6];
  const float* Bb = (const float*)d_in[7];
  const float* Cw = (const float*)d_in[8];
  const float* Cb = (const float*)d_in[9];
  const float* Uw = (const float*)d_in[10];
  const float* Ub = (const float*)d_in[11];
  const float* Vw = (const float*)d_in[12];
  const float* Vb = (const float*)d_in[13];
  const float* bnEs = (const float*)d_in[14];
  const float* bnEo = (const float*)d_in[15];
  const float* bnNs = (const float*)d_in[16];
  const float* bnNo = (const float*)d_in[17];

  float* out_h = (float*)d_out;                 // h_new [N,128]
  float* out_e = out_h + (size_t)GN * GD;       // e_new [E,128] (stages eta)

  const size_t ND = (size_t)GN * GD;
  float* ws = (float*)d_ws;
  float* hA = ws;                // [N,128]
  float* hB = hA + ND;           // [N,128]
  float* hU = hB + ND;           // [N,128]
  float* hV = hU + ND;           // [N,128]
  float* w_sum = hV + ND;        // [N,128]  (zeroed)
  float* agg = w_sum + ND;       // [N,128]  (zeroed)
  float* esum = agg + ND;        // [128]    (zeroed)
  float* esumsq = esum + GD;     // [128]    (zeroed)
  float* nsum = esumsq + GD;     // [128]    (zeroed)
  float* nsumsq = nsum + GD;     // [128]    (zeroed)
  float* ebn = nsumsq + GD;      // [256] g|b
  float* nbn = ebn + 2 * GD;     // [256] g|b

  // zero accumulators (contiguous region), capturable async memset
  hipMemsetAsync(w_sum, 0, (2 * ND + 4 * GD) * sizeof(float), stream);

  node_gemm4<<<(GN / 16 + WPB - 1) / WPB, TPB, 0, stream>>>(
      h, Aw, Ab, Bw, Bb, Uw, Ub, Vw, Vb, hA, hB, hU, hV);
  edge_gemm<<<(GE / 16) / WPB, TPB, 0, stream>>>(
      e, Cw, Cb, hA, hB, senders, receivers, out_e, esum, esumsq);
  bn_finalize<<<1, GD, 0, stream>>>(esum, esumsq, bnEs, bnEo, 1.f / GE, ebn);
  edge_apply<<<(GE * GD) / 256, 256, 0, stream>>>(
      e, ebn, senders, receivers, hV, out_e, w_sum, agg);
  node_pre<<<1024, 256, 0, stream>>>(hU, agg, w_sum, out_h, nsum, nsumsq);
  bn_finalize<<<1, GD, 0, stream>>>(nsum, nsumsq, bnNs, bnNo, 1.f / GN, nbn);
  node_apply<<<(GN * GD) / 256, 256, 0, stream>>>(h, nbn, out_h);
}